// LocalAttention_25623774888520
// MI455X (gfx1250) — compile-verified
//
#include <hip/hip_runtime.h>

typedef float v2f __attribute__((ext_vector_type(2)));
typedef float v4f __attribute__((ext_vector_type(4)));
typedef float v8f __attribute__((ext_vector_type(8)));

#define N_TOK     4096
#define C_DIM     384
#define QKV_DIM   1152
#define NUM_HEADS 8
#define HEAD_DIM  48
#define WIN       7
#define HALF      3

// C[M,N] = A[M,K] @ B[K,N] (+ bias), fp32 via V_WMMA_F32_16X16X4_F32.
// One wave computes a 64x64 output tile: 4 M-subtiles x 4 N-subtiles.
// Per k-step (K += 4): 4 A-fragments (b64 each) + 4 B-fragments (2 x b32 each)
// feed 16 WMMAs -> 32768 FLOP / 2048 B = 16 FLOP/byte from cache.
__global__ __launch_bounds__(128)
void wmma_gemm_f32(const float* __restrict__ A, const float* __restrict__ B,
                   const float* __restrict__ bias, float* __restrict__ C,
                   int M, int N, int K)
{
    const int lane = threadIdx.x & 31;
    const int wave = threadIdx.x >> 5;

    const int m0 = (blockIdx.y * 4 + wave) * 64;   // 64-row tile per wave
    const int n0 = blockIdx.x * 64;                // 64-col tile

    const int lm    = lane & 15;                   // M within subtile (A) / N within subtile (B,C)
    const int khalf = (lane >> 4) << 1;            // lanes 0-15: K=k,k+1 ; 16-31: K=k+2,k+3

    // A fragment pointers: one per M-subtile, advanced by 4 floats per k-step.
    const float* Ap0 = A + (size_t)(m0 + lm +  0) * K + khalf;
    const float* Ap1 = A + (size_t)(m0 + lm + 16) * K + khalf;
    const float* Ap2 = A + (size_t)(m0 + lm + 32) * K + khalf;
    const float* Ap3 = A + (size_t)(m0 + lm + 48) * K + khalf;
    // B fragment pointer: row kb, advanced by 4 rows per k-step.
    const float* Bp  = B + (size_t)khalf * N + n0 + lm;

    v8f acc[4][4];
    #pragma unroll
    for (int mt = 0; mt < 4; ++mt)
        #pragma unroll
        for (int nt = 0; nt < 4; ++nt)
            acc[mt][nt] = v8f{};

    for (int k = 0; k < K; k += 4) {
        v2f a0 = *(const v2f*)Ap0;
        v2f a1 = *(const v2f*)Ap1;
        v2f a2 = *(const v2f*)Ap2;
        v2f a3 = *(const v2f*)Ap3;
        Ap0 += 4; Ap1 += 4; Ap2 += 4; Ap3 += 4;

        #pragma unroll
        for (int nt = 0; nt < 4; ++nt) {
            v2f b;
            b.x = Bp[nt * 16];
            b.y = Bp[nt * 16 + N];
            acc[0][nt] = __builtin_amdgcn_wmma_f32_16x16x4_f32(false, a0, false, b, (short)0, acc[0][nt], false, false);
            acc[1][nt] = __builtin_amdgcn_wmma_f32_16x16x4_f32(false, a1, false, b, (short)0, acc[1][nt], false, false);
            acc[2][nt] = __builtin_amdgcn_wmma_f32_16x16x4_f32(false, a2, false, b, (short)0, acc[2][nt], false, false);
            acc[3][nt] = __builtin_amdgcn_wmma_f32_16x16x4_f32(false, a3, false, b, (short)0, acc[3][nt], false, false);
        }
        Bp += (size_t)4 * N;
    }

    // C/D layout: VGPR r -> lanes 0-15: M=r, lanes 16-31: M=8+r, N=lane%16
    const int crow = (lane >> 4) * 8;
    #pragma unroll
    for (int nt = 0; nt < 4; ++nt) {
        const int n = n0 + nt * 16 + lm;
        const float bv = bias ? bias[n] : 0.0f;
        #pragma unroll
        for (int mt = 0; mt < 4; ++mt) {
            float* Cp = C + (size_t)(m0 + mt * 16 + crow) * N + n;
            #pragma unroll
            for (int r = 0; r < 8; ++r)
                Cp[(size_t)r * N] = acc[mt][nt][r] + bv;
        }
    }
}

// One thread per (head, token). qkv layout: [token][3*C]: q at +0, k at +C, v at +2C;
// head h occupies dims [h*48, h*48+48), 16B-aligned -> b128 loads.
// Zero-padded (unmasked) 7x7 window: OOB entries score exactly 0 (still in softmax),
// contribute zero value.
__global__ __launch_bounds__(256)
void local_attn(const float* __restrict__ qkv, float* __restrict__ out)
{
    const int gid = blockIdx.x * blockDim.x + threadIdx.x;
    const int h = gid >> 12;            // / 4096
    const int n = gid & (N_TOK - 1);
    const int t  = n >> 10;
    const int yy = (n >> 5) & 31;
    const int xx = n & 31;

    const float scale = 0.14433756729740643f;  // 48^-0.5

    v4f q[12];
    const v4f* qp = (const v4f*)(qkv + (size_t)n * QKV_DIM + h * HEAD_DIM);
    #pragma unroll
    for (int d = 0; d < 12; ++d) q[d] = qp[d];

    float s[WIN * WIN];
    int idx = 0;
    for (int dy = -HALF; dy <= HALF; ++dy) {
        for (int dx = -HALF; dx <= HALF; ++dx) {
            const int y2 = yy + dy, x2 = xx + dx;
            float sc = 0.0f;
            if (y2 >= 0 && y2 < 32 && x2 >= 0 && x2 < 32) {
                const int nb = (t << 10) + (y2 << 5) + x2;
                const v4f* kp = (const v4f*)(qkv + (size_t)nb * QKV_DIM + C_DIM + h * HEAD_DIM);
                v4f acc4 = v4f{};
                #pragma unroll
                for (int d = 0; d < 12; ++d) {
                    v4f kv = kp[d];
                    acc4.x = fmaf(q[d].x, kv.x, acc4.x);
                    acc4.y = fmaf(q[d].y, kv.y, acc4.y);
                    acc4.z = fmaf(q[d].z, kv.z, acc4.z);
                    acc4.w = fmaf(q[d].w, kv.w, acc4.w);
                }
                sc = (acc4.x + acc4.y + acc4.z + acc4.w) * scale;
            }
            s[idx++] = sc;
        }
    }

    float m = s[0];
    #pragma unroll
    for (int i = 1; i < WIN * WIN; ++i) m = fmaxf(m, s[i]);
    float sum = 0.0f;
    #pragma unroll
    for (int i = 0; i < WIN * WIN; ++i) { s[i] = __expf(s[i] - m); sum += s[i]; }
    const float inv = 1.0f / sum;

    v4f o[12];
    #pragma unroll
    for (int d = 0; d < 12; ++d) o[d] = v4f{};

    idx = 0;
    for (int dy = -HALF; dy <= HALF; ++dy) {
        for (int dx = -HALF; dx <= HALF; ++dx) {
            const int y2 = yy + dy, x2 = xx + dx;
            if (y2 >= 0 && y2 < 32 && x2 >= 0 && x2 < 32) {
                const int nb = (t << 10) + (y2 << 5) + x2;
                const v4f* vp = (const v4f*)(qkv + (size_t)nb * QKV_DIM + 2 * C_DIM + h * HEAD_DIM);
                const float w = s[idx] * inv;
                #pragma unroll
                for (int d = 0; d < 12; ++d) {
                    v4f vv = vp[d];
                    o[d].x = fmaf(w, vv.x, o[d].x);
                    o[d].y = fmaf(w, vv.y, o[d].y);
                    o[d].z = fmaf(w, vv.z, o[d].z);
                    o[d].w = fmaf(w, vv.w, o[d].w);
                }
            }
            ++idx;
        }
    }

    v4f* op = (v4f*)(out + (size_t)n * C_DIM + h * HEAD_DIM);
    #pragma unroll
    for (int d = 0; d < 12; ++d) op[d] = o[d];
}

extern "C" void kernel_launch(void* const* d_in, const int* in_sizes, int n_in,
                              void* d_out, int out_size, void* d_ws, size_t ws_size,
                              hipStream_t stream)
{
    (void)in_sizes; (void)n_in; (void)out_size; (void)ws_size;

    const float* x      = (const float*)d_in[0];
    const float* w_qkv  = (const float*)d_in[1];
    const float* w_proj = (const float*)d_in[2];
    const float* b_proj = (const float*)d_in[3];
    float* out = (float*)d_out;

    float* qkv      = (float*)d_ws;                          // 4096 x 1152
    float* attn_out = qkv + (size_t)N_TOK * QKV_DIM;         // 4096 x 384

    // 1) qkv = x @ w_qkv          M=4096, N=1152, K=384
    {
        dim3 grid(QKV_DIM / 64, N_TOK / 256);                // (18, 16)
        wmma_gemm_f32<<<grid, 128, 0, stream>>>(x, w_qkv, nullptr, qkv,
                                                N_TOK, QKV_DIM, C_DIM);
    }

    // 2) windowed attention per (head, token)
    local_attn<<<(NUM_HEADS * N_TOK) / 256, 256, 0, stream>>>(qkv, attn_out);

    // 3) out = attn_out @ w_proj + b_proj   M=4096, N=384, K=384
    {
        dim3 grid(C_DIM / 64, N_TOK / 256);                  // (6, 16)
        wmma_gemm_f32<<<grid, 128, 0, stream>>>(attn_out, w_proj, b_proj, out,
                                                N_TOK, C_DIM, C_DIM);
    }
}